// CMPN_45964740002210
// MI455X (gfx1250) — compile-verified
//
#include <hip/hip_runtime.h>
#include <math.h>

typedef __attribute__((ext_vector_type(2))) float v2f;
typedef __attribute__((ext_vector_type(8))) float v8f;

static __device__ __forceinline__ v8f wmma4(v2f a, v2f b, v8f c) {
  // D = A(16x4 f32) * B(4x16 f32) + C(16x16 f32), full fp32 matrix op
  return __builtin_amdgcn_wmma_f32_16x16x4_f32(false, a, false, b, (short)0, c, false, false);
}

template<bool HAS_ADD, bool HAS_BIAS, bool RELU, bool GUARD>
static __device__ __forceinline__ void store_tile(
    float* __restrict__ out, const float* __restrict__ addm,
    const float* __restrict__ bias, v8f acc, int row0, int col, int M, int N)
{
#pragma unroll
  for (int g = 0; g < 8; ++g) {
    const int row = row0 + g;
    if (!GUARD || row < M) {
      float v = acc[g];
      if (HAS_ADD)  v += addm[(long)row * N + col];
      if (HAS_BIAS) v += bias[col];
      if (RELU)     v = fmaxf(v, 0.f);
      out[(long)row * N + col] = v;
    }
  }
}

// out[M,N] = act( X[M,K] @ W[N,K]^T (+ addm[M,N]) (+ bias[N]) )
// One wave computes a 32x64 tile: 2 A-fragments x 4 B-fragments = 8 WMMA / k-step.
// Hot loop is branch-free: row indices clamped (loads always valid), K-tail handled
// once after the loop with value masks on A only.
template<bool HAS_ADD, bool HAS_BIAS, bool RELU>
__global__ __launch_bounds__(256) void gemm_xwt(
    const float* __restrict__ X, const float* __restrict__ W,
    const float* __restrict__ addm, const float* __restrict__ bias,
    float* __restrict__ out, int M, int N, int K)
{
  const int cs = N >> 6;                 // 64-col strips
  const int rt = (M + 31) >> 5;          // 32-row tiles
  const int wid = blockIdx.x * (blockDim.x >> 5) + (threadIdx.x >> 5);
  if (wid >= rt * cs) return;            // wave-uniform exit: EXEC all-1s below
  const int row_t = wid / cs;
  const int strip = wid - row_t * cs;
  const int i0 = row_t << 5;
  const int j0 = strip << 6;
  const int lane = threadIdx.x & 31;
  const int r  = lane & 15;              // A row / B col / D col within 16x16 tile
  const int kh = (lane >> 4) << 1;       // K sub-offset: lanes 0-15 -> {0,1}, 16-31 -> {2,3}
  const int Mm1 = M - 1;
  const int ra0 = min(i0 + r,      Mm1); // clamped: loads always in-bounds
  const int ra1 = min(i0 + 16 + r, Mm1);

  const float* xp0 = X + (long)ra0 * K + kh;
  const float* xp1 = X + (long)ra1 * K + kh;
  const float* wq0 = W + (long)(j0 + r) * K + kh;
  const float* wq1 = wq0 + (long)16 * K;
  const float* wq2 = wq0 + (long)32 * K;
  const float* wq3 = wq0 + (long)48 * K;
  __builtin_prefetch(wq0, 0, 1);
  __builtin_prefetch(wq1, 0, 1);
  __builtin_prefetch(wq2, 0, 1);
  __builtin_prefetch(wq3, 0, 1);

  v8f a00 = {}, a01 = {}, a02 = {}, a03 = {};
  v8f a10 = {}, a11 = {}, a12 = {}, a13 = {};
  const int Km = K & ~3;
  for (int k0 = 0; k0 < Km; k0 += 4) {
    v2f fa0; fa0.x = xp0[k0]; fa0.y = xp0[k0 + 1];
    v2f fa1; fa1.x = xp1[k0]; fa1.y = xp1[k0 + 1];
    v2f fb0; fb0.x = wq0[k0]; fb0.y = wq0[k0 + 1];
    v2f fb1; fb1.x = wq1[k0]; fb1.y = wq1[k0 + 1];
    v2f fb2; fb2.x = wq2[k0]; fb2.y = wq2[k0 + 1];
    v2f fb3; fb3.x = wq3[k0]; fb3.y = wq3[k0 + 1];
    a00 = wmma4(fa0, fb0, a00); a01 = wmma4(fa0, fb1, a01);
    a02 = wmma4(fa0, fb2, a02); a03 = wmma4(fa0, fb3, a03);
    a10 = wmma4(fa1, fb0, a10); a11 = wmma4(fa1, fb1, a11);
    a12 = wmma4(fa1, fb2, a12); a13 = wmma4(fa1, fb3, a13);
  }
  if (Km < K) {                          // uniform K-tail (K % 4 != 0)
    const float m0 = (Km + kh     < K) ? 1.f : 0.f;
    const float m1 = (Km + kh + 1 < K) ? 1.f : 0.f;
    const int   o0 = min(Km + kh,     K - 1) - kh;  // clamped safe offsets
    const int   o1 = min(Km + kh + 1, K - 1) - kh;
    v2f fa0; fa0.x = xp0[o0] * m0; fa0.y = xp0[o1] * m1;
    v2f fa1; fa1.x = xp1[o0] * m0; fa1.y = xp1[o1] * m1;
    v2f fb0; fb0.x = wq0[o0]; fb0.y = wq0[o1];  // B unmasked: A zeros kill products
    v2f fb1; fb1.x = wq1[o0]; fb1.y = wq1[o1];
    v2f fb2; fb2.x = wq2[o0]; fb2.y = wq2[o1];
    v2f fb3; fb3.x = wq3[o0]; fb3.y = wq3[o1];
    a00 = wmma4(fa0, fb0, a00); a01 = wmma4(fa0, fb1, a01);
    a02 = wmma4(fa0, fb2, a02); a03 = wmma4(fa0, fb3, a03);
    a10 = wmma4(fa1, fb0, a10); a11 = wmma4(fa1, fb1, a11);
    a12 = wmma4(fa1, fb2, a12); a13 = wmma4(fa1, fb3, a13);
  }

  const int mh = (lane >> 4) << 3;       // D: VGPR g -> row g / g+8 per lane half
  const int r0 = i0 + mh, r1 = i0 + 16 + mh;
  const int c0 = j0 + r;
  if (i0 + 31 < M) {                     // wave-uniform fast path: no store guards
    store_tile<HAS_ADD,HAS_BIAS,RELU,false>(out, addm, bias, a00, r0, c0,      M, N);
    store_tile<HAS_ADD,HAS_BIAS,RELU,false>(out, addm, bias, a01, r0, c0 + 16, M, N);
    store_tile<HAS_ADD,HAS_BIAS,RELU,false>(out, addm, bias, a02, r0, c0 + 32, M, N);
    store_tile<HAS_ADD,HAS_BIAS,RELU,false>(out, addm, bias, a03, r0, c0 + 48, M, N);
    store_tile<HAS_ADD,HAS_BIAS,RELU,false>(out, addm, bias, a10, r1, c0,      M, N);
    store_tile<HAS_ADD,HAS_BIAS,RELU,false>(out, addm, bias, a11, r1, c0 + 16, M, N);
    store_tile<HAS_ADD,HAS_BIAS,RELU,false>(out, addm, bias, a12, r1, c0 + 32, M, N);
    store_tile<HAS_ADD,HAS_BIAS,RELU,false>(out, addm, bias, a13, r1, c0 + 48, M, N);
  } else {
    store_tile<HAS_ADD,HAS_BIAS,RELU,true>(out, addm, bias, a00, r0, c0,      M, N);
    store_tile<HAS_ADD,HAS_BIAS,RELU,true>(out, addm, bias, a01, r0, c0 + 16, M, N);
    store_tile<HAS_ADD,HAS_BIAS,RELU,true>(out, addm, bias, a02, r0, c0 + 32, M, N);
    store_tile<HAS_ADD,HAS_BIAS,RELU,true>(out, addm, bias, a03, r0, c0 + 48, M, N);
    store_tile<HAS_ADD,HAS_BIAS,RELU,true>(out, addm, bias, a10, r1, c0,      M, N);
    store_tile<HAS_ADD,HAS_BIAS,RELU,true>(out, addm, bias, a11, r1, c0 + 16, M, N);
    store_tile<HAS_ADD,HAS_BIAS,RELU,true>(out, addm, bias, a12, r1, c0 + 32, M, N);
    store_tile<HAS_ADD,HAS_BIAS,RELU,true>(out, addm, bias, a13, r1, c0 + 48, M, N);
  }
}

// message_atom += sum(nei)*max(nei) over 6 gathered bond messages
__global__ void k_agg_step(const int* __restrict__ a2b, const float* __restrict__ mbond,
                           float* __restrict__ matom, int n_atoms)
{
  int idx = blockIdx.x * blockDim.x + threadIdx.x;
  if (idx >= n_atoms * 256) return;
  int a = idx >> 8, j = idx & 255;
  const int* nb = a2b + a * 6;
  float s = 0.f, mx = -__builtin_inff();
#pragma unroll
  for (int t = 0; t < 6; ++t) {
    float v = mbond[(nb[t] << 8) + j];
    s += v; mx = fmaxf(mx, v);
  }
  matom[idx] += s * mx;
}

// cat = [sum*max agg | message_atom | input_atom]  (16385 x 768)
__global__ void k_agg_cat(const int* __restrict__ a2b, const float* __restrict__ mbond,
                          const float* __restrict__ matom, const float* __restrict__ iatom,
                          float* __restrict__ cat, int n_atoms)
{
  int idx = blockIdx.x * blockDim.x + threadIdx.x;
  if (idx >= n_atoms * 256) return;
  int a = idx >> 8, j = idx & 255;
  const int* nb = a2b + a * 6;
  float s = 0.f, mx = -__builtin_inff();
#pragma unroll
  for (int t = 0; t < 6; ++t) {
    float v = mbond[(nb[t] << 8) + j];
    s += v; mx = fmaxf(mx, v);
  }
  float* c = cat + a * 768;
  c[j] = s * mx;
  c[256 + j] = matom[idx];
  c[512 + j] = iatom[idx];
}

// mbtmp = message_atom[b2a] - message_bond[b2revb]
__global__ void k_bond_gather(const int* __restrict__ b2a, const int* __restrict__ b2revb,
                              const float* __restrict__ matom, const float* __restrict__ mbond,
                              float* __restrict__ mbtmp, int n_bonds)
{
  int idx = blockIdx.x * blockDim.x + threadIdx.x;
  if (idx >= n_bonds * 256) return;
  int b = idx >> 8, j = idx & 255;
  mbtmp[idx] = matom[(b2a[b] << 8) + j] - mbond[(b2revb[b] << 8) + j];
}

// msg = relu(agg + gru_bias); also agg2 row 0 = [msg0 | msg0]
__global__ void k_msg(const float* __restrict__ agg, const float* __restrict__ gbias,
                      float* __restrict__ msg, float* __restrict__ agg2, int n_atoms)
{
  int idx = blockIdx.x * blockDim.x + threadIdx.x;
  if (idx >= n_atoms * 256) return;
  int i = idx >> 8, j = idx & 255;
  float v = fmaxf(agg[idx] + gbias[j], 0.f);
  msg[idx] = v;
  if (i == 0) { agg2[j] = v; agg2[256 + j] = v; }
}

// h0[b][j] = max_s agg[1 + b*64 + s][j]
__global__ void k_h0(const float* __restrict__ agg, float* __restrict__ h0)
{
  int idx = blockIdx.x * blockDim.x + threadIdx.x;   // 256*256
  int b = idx >> 8, j = idx & 255;
  const float* p = agg + 256 + b * (64 * 256) + j;
  float m = -__builtin_inff();
  for (int s = 0; s < 64; ++s) m = fmaxf(m, p[s * 256]);
  h0[idx] = m;
}

// Persistent GRU scan: block owns 16 molecules, h & hW in LDS, 64 sequential steps.
// Each step: hW[16x768] = h[16x256] @ Whh^T via WMMA, then fused gate math.
template<int DIR>
__global__ __launch_bounds__(256) void k_gru_scan(
    const float* __restrict__ xW,   // [16384][768] (r,z,n), rows b*64+s
    const float* __restrict__ Whh,  // [768][256]
    const float* __restrict__ bhh,  // [768]
    const float* __restrict__ h0,   // [256][256]
    float* __restrict__ agg2)       // [16385][512]
{
  __shared__ float h_lds[16][256];   // 16 KB
  __shared__ float hw_lds[16][768];  // 48 KB
  const int mol0 = blockIdx.x << 4;
  const int tid = threadIdx.x;
  for (int e = tid; e < 16 * 256; e += 256)
    h_lds[e >> 8][e & 255] = h0[(mol0 << 8) + e];
  __syncthreads();

  const int wave = tid >> 5, lane = tid & 31;
  const int r = lane & 15, kh = (lane >> 4) << 1;
  const int mh = (lane >> 4) << 3;
  const int jbase = wave * 96;       // 8 waves x 6 col-tiles = 768 cols

  const float* hp  = &h_lds[r][kh];
  const float* wq0 = Whh + (jbase + r) * 256 + kh;
  const float* wq1 = wq0 + 16 * 256;
  const float* wq2 = wq0 + 32 * 256;
  const float* wq3 = wq0 + 48 * 256;
  const float* wq4 = wq0 + 64 * 256;
  const float* wq5 = wq0 + 80 * 256;

  for (int t = 0; t < 64; ++t) {
    const int pos = DIR ? (63 - t) : t;
    v8f c0 = {}, c1 = {}, c2 = {}, c3 = {}, c4 = {}, c5 = {};
    for (int k0 = 0; k0 < 256; k0 += 4) {
      v2f fa; fa.x = hp[k0]; fa.y = hp[k0 + 1];
      v2f b0; b0.x = wq0[k0]; b0.y = wq0[k0 + 1];
      v2f b1; b1.x = wq1[k0]; b1.y = wq1[k0 + 1];
      v2f b2; b2.x = wq2[k0]; b2.y = wq2[k0 + 1];
      v2f b3; b3.x = wq3[k0]; b3.y = wq3[k0 + 1];
      v2f b4; b4.x = wq4[k0]; b4.y = wq4[k0 + 1];
      v2f b5; b5.x = wq5[k0]; b5.y = wq5[k0 + 1];
      c0 = wmma4(fa, b0, c0); c1 = wmma4(fa, b1, c1); c2 = wmma4(fa, b2, c2);
      c3 = wmma4(fa, b3, c3); c4 = wmma4(fa, b4, c4); c5 = wmma4(fa, b5, c5);
    }
#pragma unroll
    for (int g = 0; g < 8; ++g) {
      hw_lds[mh + g][jbase +  0 + r] = c0[g];
      hw_lds[mh + g][jbase + 16 + r] = c1[g];
      hw_lds[mh + g][jbase + 32 + r] = c2[g];
      hw_lds[mh + g][jbase + 48 + r] = c3[g];
      hw_lds[mh + g][jbase + 64 + r] = c4[g];
      hw_lds[mh + g][jbase + 80 + r] = c5[g];
    }
    __syncthreads();

    for (int e = tid; e < 16 * 256; e += 256) {
      const int m = e >> 8, j = e & 255;
      const int row = ((mol0 + m) << 6) + pos;
      const float* xp = xW + row * 768;
      const float xr = xp[j], xz = xp[256 + j], xn = xp[512 + j];
      const float hr = hw_lds[m][j]       + bhh[j];
      const float hz = hw_lds[m][256 + j] + bhh[256 + j];
      const float hn = hw_lds[m][512 + j] + bhh[512 + j];
      const float rr = 1.f / (1.f + __expf(-(xr + hr)));
      const float zz = 1.f / (1.f + __expf(-(xz + hz)));
      const float nn = tanhf(xn + rr * hn);
      const float hv = (1.f - zz) * nn + zz * h_lds[m][j];
      h_lds[m][j] = hv;
      agg2[(long)(row + 1) * 512 + DIR * 256 + j] = hv;
    }
    __syncthreads();
  }
}

extern "C" void kernel_launch(void* const* d_in, const int* in_sizes, int n_in,
                              void* d_out, int out_size, void* d_ws, size_t ws_size,
                              hipStream_t stream)
{
  const float* f_atoms  = (const float*)d_in[0];
  const float* f_bonds  = (const float*)d_in[1];
  const float* W_i_atom = (const float*)d_in[2];
  const float* W_i_bond = (const float*)d_in[3];
  const float* W_h      = (const float*)d_in[4];
  const float* lr_W     = (const float*)d_in[5];
  const float* W_o_W    = (const float*)d_in[6];
  const float* W_o_b    = (const float*)d_in[7];
  const float* gru_bias = (const float*)d_in[8];
  const float* Wih_f    = (const float*)d_in[9];
  const float* Whh_f    = (const float*)d_in[10];
  const float* bih_f    = (const float*)d_in[11];
  const float* bhh_f    = (const float*)d_in[12];
  const float* Wih_b    = (const float*)d_in[13];
  const float* Whh_b    = (const float*)d_in[14];
  const float* bih_b    = (const float*)d_in[15];
  const float* bhh_b    = (const float*)d_in[16];
  const int* a2b    = (const int*)d_in[17];
  const int* b2a    = (const int*)d_in[18];
  const int* b2revb = (const int*)d_in[19];
  (void)in_sizes; (void)n_in; (void)out_size; (void)ws_size;

  const int NA = 16385, NBD = 65537;

  float* ws = (float*)d_ws;
  size_t o = 0;
  float* input_atom = ws + o; o += (size_t)NA  * 256;
  float* input_bond = ws + o; o += (size_t)NBD * 256;
  float* matom      = ws + o; o += (size_t)NA  * 256;
  float* mbond      = ws + o; o += (size_t)NBD * 256;
  float* mbtmp      = ws + o; o += (size_t)NBD * 256;
  float* cat        = ws + o; o += (size_t)NA  * 768;
  float* aggm       = ws + o; o += (size_t)NA  * 256;
  float* msg        = ws + o; o += (size_t)NA  * 256;
  float* h0         = ws + o; o += (size_t)256 * 256;
  float* xWf        = ws + o; o += (size_t)16384 * 768;
  float* xWb        = ws + o; o += (size_t)16384 * 768;
  float* agg2       = ws + o; o += (size_t)NA  * 512;

  auto gemm_grid = [](int M, int N) {
    int waves = ((M + 31) >> 5) * (N >> 6);
    return (waves + 7) / 8;
  };
  const int eb_atom = (NA  * 256 + 255) / 256;
  const int eb_bond = (NBD * 256 + 255) / 256;

  // input projections
  gemm_xwt<false,false,true><<<gemm_grid(NA, 256), 256, 0, stream>>>(
      f_atoms, W_i_atom, nullptr, nullptr, input_atom, NA, 256, 133);
  gemm_xwt<false,false,true><<<gemm_grid(NBD, 256), 256, 0, stream>>>(
      f_bonds, W_i_bond, nullptr, nullptr, input_bond, NBD, 256, 147);
  hipMemcpyAsync(matom, input_atom, (size_t)NA  * 256 * sizeof(float),
                 hipMemcpyDeviceToDevice, stream);
  hipMemcpyAsync(mbond, input_bond, (size_t)NBD * 256 * sizeof(float),
                 hipMemcpyDeviceToDevice, stream);

  // depth loop
  for (int d = 0; d < 5; ++d) {
    k_agg_step<<<eb_atom, 256, 0, stream>>>(a2b, mbond, matom, NA);
    k_bond_gather<<<eb_bond, 256, 0, stream>>>(b2a, b2revb, matom, mbond, mbtmp, NBD);
    gemm_xwt<true,false,true><<<gemm_grid(NBD, 256), 256, 0, stream>>>(
        mbtmp, W_h + (size_t)d * 256 * 256, input_bond, nullptr, mbond, NBD, 256, 256);
  }

  // readout: cat -> lr_W -> GRU
  k_agg_cat<<<eb_atom, 256, 0, stream>>>(a2b, mbond, matom, input_atom, cat, NA);
  gemm_xwt<false,false,false><<<gemm_grid(NA, 256), 256, 0, stream>>>(
      cat, lr_W, nullptr, nullptr, aggm, NA, 256, 768);
  k_msg<<<eb_atom, 256, 0, stream>>>(aggm, gru_bias, msg, agg2, NA);
  k_h0<<<256, 256, 0, stream>>>(aggm, h0);

  gemm_xwt<false,true,false><<<gemm_grid(16384, 768), 256, 0, stream>>>(
      msg + 256, Wih_f, nullptr, bih_f, xWf, 16384, 768, 256);
  gemm_xwt<false,true,false><<<gemm_grid(16384, 768), 256, 0, stream>>>(
      msg + 256, Wih_b, nullptr, bih_b, xWb, 16384, 768, 256);

  k_gru_scan<0><<<16, 256, 0, stream>>>(xWf, Whh_f, bhh_f, h0, agg2);
  k_gru_scan<1><<<16, 256, 0, stream>>>(xWb, Whh_b, bhh_b, h0, agg2);

  // output projection
  gemm_xwt<false,true,true><<<gemm_grid(NA, 256), 256, 0, stream>>>(
      agg2, W_o_W, nullptr, W_o_b, (float*)d_out, NA, 256, 512);
}